// FieldGeometryCoupling_43439299232032
// MI455X (gfx1250) — compile-verified
//
#include <hip/hip_runtime.h>

typedef __attribute__((ext_vector_type(2))) float v2f;
typedef __attribute__((ext_vector_type(8))) float v8f;
typedef __attribute__((ext_vector_type(4))) unsigned int u32x4;
typedef __attribute__((ext_vector_type(8))) int i32x8;
typedef __attribute__((ext_vector_type(4))) int i32x4;

// ---------------- constants from the reference ----------------
#define C0c     0.02795084971874737f    // sqrt(1/1280)
#define C1c     0.05103103630798288f    // sqrt(3/1152)
#define C2c     0.09882117688026186f    // sqrt(5/512)
#define INV_S3c 0.57735026918962576f
#define INV_S5c 0.44721359549995794f
#define RL0c    0.17677669529663687f    // 1/sqrt(32)
#define RL1c    0.25f                   // 1/sqrt(16)
#define RL2c    0.35355339059327373f    // 1/sqrt(8)

__device__ __forceinline__ float w3j112(int i, int j, int m) {
    const float s2 = 0.70710678118654752f;   // 1/sqrt(2)
    const float s6 = 0.40824829046386302f;   // 1/sqrt(6)
    const float is5 = 0.44721359549995794f;  // overall 1/sqrt(5)
    float v = 0.f;
    if      (m == 0) { if ((i==0&&j==2)||(i==2&&j==0)) v = s2; }
    else if (m == 1) { if ((i==0&&j==1)||(i==1&&j==0)) v = s2; }
    else if (m == 2) { if (i==1&&j==1) v = 2.f*s6; else if (i==0&&j==0) v = -s6; else if (i==2&&j==2) v = -s6; }
    else if (m == 3) { if ((i==1&&j==2)||(i==2&&j==1)) v = s2; }
    else             { if (i==2&&j==2) v = s2; else if (i==0&&j==0) v = -s2; }
    return v * is5;
}

// Build the fused 120x120 map W (padded to N=128, zero cols 120..127),
// stored directly in WMMA B-fragment swizzled order:
//   blob[((ntile*30 + kstep)*32 + lane)*2 + r]
// with k = 4*kstep + 2*(lane/16) + r, n = ntile*16 + lane%16.
__global__ void build_W(const float* __restrict__ f,      // 80 floats: f0[0:32], f1[v][i]=f[32+3v+i]
                        const float* __restrict__ w000,   // (32,32,32)
                        const float* __restrict__ w011,   // (32,16,16)
                        const float* __restrict__ w101,   // (16,32,16)
                        const float* __restrict__ w110,   // (16,16,32)
                        const float* __restrict__ w112,   // (16,16,8)
                        const float* __restrict__ w202,   // (8,32,8)
                        const float* __restrict__ w211,   // (8,16,16)
                        const float* __restrict__ wl0,    // (32,32)
                        const float* __restrict__ wl1,    // (16,16)
                        const float* __restrict__ wl2,    // (8,8)
                        float* __restrict__ blob) {
    int id = blockIdx.x * 256 + threadIdx.x;     // 120*128 entries
    if (id >= 120 * 128) return;
    int k = id >> 7;       // 0..119  (input feature index)
    int n = id & 127;      // 0..127  (output feature index, padded)

    float val = 0.f;
    if (n < 120) {
        if (k < 32) {                              // g0 channel u=k
            int u = k;
            if (n < 32) {                          // -> out0[w]
                int w = n; float s = 0.f;
                for (int v = 0; v < 32; ++v) s += w000[(u*32+v)*32 + w] * f[v];
                val = C0c * s + RL0c * wl0[u*32 + w];
            } else if (n < 80) {                   // -> out1[w][j]
                int w = (n-32)/3, j = (n-32)%3; float s = 0.f;
                for (int v = 0; v < 16; ++v) s += w011[(u*16+v)*16 + w] * f[32 + 3*v + j];
                val = C1c * INV_S3c * s;
            }
        } else if (k < 80) {                       // g1 channel u,i
            int u = (k-32)/3, i = (k-32)%3;
            if (n < 32) {                          // -> out0[w]
                int w = n; float s = 0.f;
                for (int v = 0; v < 16; ++v) s += w110[(u*16+v)*32 + w] * f[32 + 3*v + i];
                val = C0c * INV_S3c * s;
            } else if (n < 80) {                   // -> out1[w][j], diagonal in i==j
                int w = (n-32)/3, j = (n-32)%3;
                if (i == j) {
                    float s = 0.f;
                    for (int v = 0; v < 32; ++v) s += w101[(u*32+v)*16 + w] * f[v];
                    val = C1c * INV_S3c * s + RL1c * wl1[u*16 + w];
                }
            } else {                               // -> out2[w][m] via W3J
                int w = (n-80)/5, m = (n-80)%5; float s = 0.f;
                for (int j = 0; j < 3; ++j) {
                    float c = w3j112(i, j, m);
                    if (c != 0.f) {
                        float sv = 0.f;
                        for (int v = 0; v < 16; ++v) sv += w112[(u*16+v)*8 + w] * f[32 + 3*v + j];
                        s += c * sv;
                    }
                }
                val = C2c * s;
            }
        } else {                                   // g2 channel u,mk
            int u = (k-80)/5, mk = (k-80)%5;
            if (n >= 32 && n < 80) {               // -> out1[w][j] via W3J
                int w = (n-32)/3, j = (n-32)%3; float s = 0.f;
                for (int i = 0; i < 3; ++i) {
                    float c = w3j112(i, j, mk);
                    if (c != 0.f) {
                        float sv = 0.f;
                        for (int v = 0; v < 16; ++v) sv += w211[(u*16+v)*16 + w] * f[32 + 3*v + i];
                        s += c * sv;
                    }
                }
                val = C1c * s;
            } else if (n >= 80) {                  // -> out2[w][m], diagonal in m==mk
                int w = (n-80)/5, m = (n-80)%5;
                if (m == mk) {
                    float s = 0.f;
                    for (int v = 0; v < 32; ++v) s += w202[(u*32+v)*8 + w] * f[v];
                    val = C2c * INV_S5c * s + RL2c * wl2[u*8 + w];
                }
            }
        }
    }

    // swizzle into B-fragment layout
    int ntile = n >> 4;
    int lane  = (n & 15) + (((k >> 1) & 1) << 4);  // lane half encodes K half
    int kstep = k >> 2;
    int r     = k & 1;
    blob[(((ntile*30 + kstep)*32 + lane) << 1) + r] = val;
}

// OUT[z,0:120] = G[z,0:120] @ W ; one wave per 16 atoms, full 128-wide (padded) N.
// LDS plan: phase 1 = swizzled B blob (15360 floats, TDM-staged);
//           phase 2 (after K-loop) = per-wave output transpose buffer
//           (8 waves x 16 rows x 124-float padded rows = 15872 floats).
__launch_bounds__(256)
__global__ void tp_gemm(const float* __restrict__ G,
                        const float* __restrict__ blob,   // swizzled W, 15360 floats
                        float* __restrict__ OUT,
                        int n_atoms) {
    __shared__ float sB[15872];                    // 63488 bytes

    // ---- stage W blob into LDS via the Tensor Data Mover ----
#if __has_builtin(__builtin_amdgcn_tensor_load_to_lds) && __has_builtin(__builtin_amdgcn_s_wait_tensorcnt)
    if (threadIdx.x < 32) {                        // wave 0 issues one TDM op
        unsigned lds_base = (unsigned)(unsigned long long)(void*)sB;   // low 32b = LDS offset
        unsigned long long ga = (unsigned long long)blob;
        // D# group0: count=1, lds_addr, global_addr[56:0], type=2
        u32x4 g0 = { 1u,
                     lds_base,
                     (unsigned)(ga & 0xffffffffu),
                     (unsigned)((ga >> 32) & 0x1ffffffu) | (2u << 30) };
        // D# group1: data_size=4B; tensor_dim0=15360; tensor_dim1=1;
        //            tile_dim0=15360 (1-D tile); tensor_dim0_stride=15360
        i32x8 g1 = { (int)(2u << 16),          // workgroup_mask=0, data_size=2 (4B)
                     (int)0x3C000000u,          // tensor_dim0[15:0]=15360 in [31:16]
                     (int)0x00010000u,          // tensor_dim0[31:16]=0, tensor_dim1[15:0]=1
                     (int)0x3C000000u,          // tensor_dim1[31:16]=0, tile_dim0=15360
                     0,                         // tile_dim1=0, tile_dim2=0 (unused)
                     15360,                     // tensor_dim0_stride[31:0]
                     0, 0 };
        i32x4 gz4 = { 0, 0, 0, 0 };
        i32x8 gz8 = { 0, 0, 0, 0, 0, 0, 0, 0 };
        // 6-arg toolchain variant: (g0, g1, g2, g3, pad, cpol)
        __builtin_amdgcn_tensor_load_to_lds(g0, g1, gz4, gz4, gz8, 0);
        __builtin_amdgcn_s_wait_tensorcnt(0);
    }
#else
    for (int i = threadIdx.x; i < 15360/4; i += 256)
        ((float4*)sB)[i] = ((const float4*)blob)[i];
#endif
    __syncthreads();

    const int wv   = threadIdx.x >> 5;
    const int lane = threadIdx.x & 31;
    const int h    = lane >> 4;                    // K-half for A/B fragments
    const int lo   = lane & 15;

    const long m0  = ((long)blockIdx.x * 8 + wv) * 16;
    long row = m0 + lo;
    if (row >= n_atoms) row = n_atoms - 1;         // clamp loads; stores are masked
    const float* gp = G + row * 120 + 2*h;         // 8-byte aligned

    v8f acc[8];
#pragma unroll
    for (int t = 0; t < 8; ++t) acc[t] = (v8f){0.f,0.f,0.f,0.f,0.f,0.f,0.f,0.f};

#pragma unroll 2
    for (int ks = 0; ks < 30; ++ks) {
        // A fragment (16x4 f32): lane = row m (low 4b) + K-half, regs = K, K+1
        v2f a = *(const v2f*)(gp + ks*4);
#pragma unroll
        for (int t = 0; t < 8; ++t) {
            // B fragment: lane reads its own float2 -> conflict-free ds_read_b64
            v2f b = *(const v2f*)(sB + (((t*30 + ks)*32 + lane) << 1));
            acc[t] = __builtin_amdgcn_wmma_f32_16x16x4_f32(
                false, a, false, b, (short)0, acc[t], false, false);
        }
    }

    // ---- epilogue: transpose through LDS, then coalesced b128 stores ----
    __syncthreads();                               // B blob is dead; reuse sB

    // C/D layout: reg r, lane -> row m = r + 8*h, col n = 16*t + lane%16.
    // LDS row stride 124 floats: 16B-aligned rows, lane halves on disjoint banks.
    float* wbuf = sB + wv * (16 * 124);
#pragma unroll
    for (int t = 0; t < 8; ++t) {
        int n = t*16 + lo;
        if (n < 120) {
#pragma unroll
            for (int r = 0; r < 8; ++r)
                wbuf[(r + 8*h) * 124 + n] = acc[t][r];
        }
    }
    // per-wave: DS ops are in-order, reads below see this wave's own writes

    // wave's output block: rows m0..m0+15 = contiguous 1920 floats in OUT
    float* outp = OUT + m0 * 120;
    if (m0 + 16 <= (long)n_atoms) {                // common fast path: no predication
#pragma unroll
        for (int it = 0; it < 15; ++it) {
            int q = it*32 + lane;                  // float4 index within 480
            int m = q / 30, c4 = (q % 30) * 4;
            float4 vv = *(const float4*)&wbuf[m * 124 + c4];
            *(float4*)&outp[q * 4] = vv;           // 512B coalesced per instruction
        }
    } else {
        for (int it = 0; it < 15; ++it) {
            int q = it*32 + lane;
            int m = q / 30, c4 = (q % 30) * 4;
            if (m0 + m < (long)n_atoms) {
                float4 vv = *(const float4*)&wbuf[m * 124 + c4];
                *(float4*)&outp[q * 4] = vv;
            }
        }
    }
}

extern "C" void kernel_launch(void* const* d_in, const int* in_sizes, int n_in,
                              void* d_out, int out_size, void* d_ws, size_t ws_size,
                              hipStream_t stream) {
    const float* G    = (const float*)d_in[0];
    const float* f    = (const float*)d_in[1];
    const float* w000 = (const float*)d_in[2];
    const float* w011 = (const float*)d_in[3];
    const float* w101 = (const float*)d_in[4];
    const float* w110 = (const float*)d_in[5];
    const float* w112 = (const float*)d_in[6];
    const float* w202 = (const float*)d_in[7];
    const float* w211 = (const float*)d_in[8];
    const float* wl0  = (const float*)d_in[9];
    const float* wl1  = (const float*)d_in[10];
    const float* wl2  = (const float*)d_in[11];

    const int n_atoms = in_sizes[0] / 120;
    float* blob = (float*)d_ws;                    // 61440 bytes used

    build_W<<<60, 256, 0, stream>>>(f, w000, w011, w101, w110, w112, w202, w211,
                                    wl0, wl1, wl2, blob);

    const int grid = (n_atoms + 127) / 128;        // 128 atoms per block (8 waves x 16)
    tp_gemm<<<grid, 256, 0, stream>>>(G, blob, (float*)d_out, n_atoms);
}